// DSA_19499151524393
// MI455X (gfx1250) — compile-verified
//
#include <hip/hip_runtime.h>
#include <math.h>

typedef __attribute__((ext_vector_type(2))) float v2f;
typedef __attribute__((ext_vector_type(4))) float v4f;
typedef __attribute__((ext_vector_type(8))) float v8f;

constexpr int U   = 4;
constexpr int KK  = 3;          // kernel size
constexpr int D   = 8;          // U*2 hidden dim
constexpr int NU  = 32;
constexpr int C   = 256;
constexpr int THW = 8 * 28 * 28;   // 6272 (contiguous per (nu,c))
constexpr int THW4 = THW / 4;      // 1568 float4s
constexpr int N   = NU / U;        // 8 clips
constexpr int R   = N * C;         // 2048 MLP rows
constexpr float EPS = 1e-5f;

// ---------------------------------------------------------------------------
// Kernel 1: global average pool over (t,h,w) per (nu,c).
// pooled layout: pooled[(n*C + c)*U + u]  (row-major for the WMMA A-matrix).
// One 256-thread block per (nu,c); b128 loads; LDS tree reduce.
// ---------------------------------------------------------------------------
__global__ void dsa_pool_kernel(const float* __restrict__ x,
                                float* __restrict__ pooled) {
    __shared__ float red[256];
    const int flat = blockIdx.x;            // nu*C + c
    const int tid  = threadIdx.x;
    const v4f* src = reinterpret_cast<const v4f*>(x + (size_t)flat * THW);
    float s = 0.0f;
    for (int i = tid; i < THW4; i += 256) {
        v4f v = src[i];
        s += v.x + v.y + v.z + v.w;
    }
    red[tid] = s;
    __syncthreads();
    for (int off = 128; off > 0; off >>= 1) {
        if (tid < off) red[tid] += red[tid + off];
        __syncthreads();
    }
    if (tid == 0) {
        const int nu = flat / C, c = flat % C;
        const int r  = (nu / U) * C + c;
        const int u  = nu % U;
        pooled[r * U + u] = red[0] * (1.0f / (float)THW);
    }
}

// ---------------------------------------------------------------------------
// Kernel 2: kernel-generator MLP via V_WMMA_F32_16X16X4_F32.
// One wave (32 lanes) handles 16 rows of pooled (16x4).
//   GEMM1: z(16x16) = pooled(16x4) x W1t(4x16, cols>=8 zero)      [1 wmma]
//   BN+ReLU in D-layout (lane column d = lane&15)
//   LDS bounce: D-layout -> A-layout
//   GEMM2: logits = relu_z(16x8) x W2t(8x16, cols>=3 zero)        [2 wmma, K split]
//   softmax over K=3, write kern[(row)*3 + k]
// ---------------------------------------------------------------------------
__global__ void dsa_mlp_wmma_kernel(const float* __restrict__ pooled,
                                    const float* __restrict__ W1,     // (8,4)
                                    const float* __restrict__ gamma,
                                    const float* __restrict__ beta,
                                    const float* __restrict__ mean,
                                    const float* __restrict__ var,
                                    const float* __restrict__ W2,     // (3,8)
                                    float* __restrict__ kern) {
    __shared__ float zsh[16 * 16];
    const int lane = threadIdx.x;        // 0..31, full wave (EXEC all ones)
    const int base = blockIdx.x * 16;    // first row of this tile
    const int M    = lane & 15;          // A-row / B-col index for this lane
    const int hi   = lane >> 4;          // lane half
    const int K0   = hi * 2;             // K pair held by this lane half

    // --- GEMM1: A = pooled rows, B = W1^T (4x16, columns d>=8 are zero) ---
    v2f a1, b1;
    a1.x = pooled[(base + M) * U + K0];
    a1.y = pooled[(base + M) * U + K0 + 1];
    b1.x = (M < D) ? W1[M * U + K0]     : 0.0f;   // B[K0  ][d=M] = W1[d][K0]
    b1.y = (M < D) ? W1[M * U + K0 + 1] : 0.0f;   // B[K0+1][d=M]
    v8f zacc = {};
    zacc = __builtin_amdgcn_wmma_f32_16x16x4_f32(
        false, a1, false, b1, (short)0, zacc, false, false);

    // --- BN (eval) + ReLU; lane's column d = M ---
    float scale = 0.0f, shift = 0.0f;
    if (M < D) {
        const float inv = rsqrtf(var[M] + EPS);
        scale = gamma[M] * inv;
        shift = beta[M] - mean[M] * scale;
    }
#pragma unroll
    for (int v = 0; v < 8; ++v) {
        float z = fmaxf(zacc[v] * scale + shift, 0.0f);
        zsh[(v + 8 * hi) * 16 + M] = z;    // D-layout row = v + 8*hi, col = M
    }
    __syncthreads();

    // --- GEMM2: A = relu(z) 16x8 split into two 16x4 K-tiles; accumulate ---
    v2f a2lo, a2hi, b2lo, b2hi;
    a2lo.x = zsh[M * 16 + K0];
    a2lo.y = zsh[M * 16 + K0 + 1];
    a2hi.x = zsh[M * 16 + 4 + K0];
    a2hi.y = zsh[M * 16 + 4 + K0 + 1];
    // B2[d][k] = (k<3) ? W2[k][d] : 0 ; lane's column k = M
    b2lo.x = (M < KK) ? W2[M * D + K0]         : 0.0f;
    b2lo.y = (M < KK) ? W2[M * D + K0 + 1]     : 0.0f;
    b2hi.x = (M < KK) ? W2[M * D + 4 + K0]     : 0.0f;
    b2hi.y = (M < KK) ? W2[M * D + 4 + K0 + 1] : 0.0f;
    v8f lacc = {};
    lacc = __builtin_amdgcn_wmma_f32_16x16x4_f32(
        false, a2lo, false, b2lo, (short)0, lacc, false, false);
    lacc = __builtin_amdgcn_wmma_f32_16x16x4_f32(
        false, a2hi, false, b2hi, (short)0, lacc, false, false);

    __syncthreads();                       // done reading zsh as A2
#pragma unroll
    for (int v = 0; v < 8; ++v)
        zsh[(v + 8 * hi) * 16 + M] = lacc[v];
    __syncthreads();

    // --- softmax over K=3, one row per lane (lanes 0..15) ---
    if (lane < 16) {
        const float l0 = zsh[lane * 16 + 0];
        const float l1 = zsh[lane * 16 + 1];
        const float l2 = zsh[lane * 16 + 2];
        const float m  = fmaxf(l0, fmaxf(l1, l2));
        const float e0 = __expf(l0 - m);
        const float e1 = __expf(l1 - m);
        const float e2 = __expf(l2 - m);
        const float is = 1.0f / (e0 + e1 + e2);
        float* kr = kern + (size_t)(base + lane) * KK;
        kr[0] = e0 * is; kr[1] = e1 * is; kr[2] = e2 * is;
    }
}

// ---------------------------------------------------------------------------
// Kernel 3: dynamic shift-conv along U, fused across all 4 segments so every
// x element is loaded exactly once (b128) and every output stored once.
// NT stores keep the 205MB output from evicting x (x ~fits the 192MB L2,
// so this pass reads mostly from L2 warmed by the pool pass).
//   out[0] = k1*x0 + k2*x1
//   out[1] = k0*x0 + k1*x1 + k2*x2
//   out[2] = k0*x1 + k1*x2 + k2*x3
//   out[3] = k0*x2 + k1*x3
// ---------------------------------------------------------------------------
__device__ inline v4f f4fma(float a, v4f v, v4f acc) {
    acc.x = fmaf(a, v.x, acc.x);
    acc.y = fmaf(a, v.y, acc.y);
    acc.z = fmaf(a, v.z, acc.z);
    acc.w = fmaf(a, v.w, acc.w);
    return acc;
}

__global__ void dsa_conv_kernel(const float* __restrict__ x,
                                const float* __restrict__ kern,
                                float* __restrict__ out) {
    const int flat = blockIdx.x;         // n*C + c
    const int tid  = threadIdx.x;
    const int n    = flat / C;
    const int c    = flat % C;
    const float k0 = kern[flat * KK + 0];
    const float k1 = kern[flat * KK + 1];
    const float k2 = kern[flat * KK + 2];

    const size_t s0 = ((size_t)((n * U + 0) * C + c)) * THW;
    const size_t s1 = ((size_t)((n * U + 1) * C + c)) * THW;
    const size_t s2 = ((size_t)((n * U + 2) * C + c)) * THW;
    const size_t s3 = ((size_t)((n * U + 3) * C + c)) * THW;

    const v4f* x0 = reinterpret_cast<const v4f*>(x + s0);
    const v4f* x1 = reinterpret_cast<const v4f*>(x + s1);
    const v4f* x2 = reinterpret_cast<const v4f*>(x + s2);
    const v4f* x3 = reinterpret_cast<const v4f*>(x + s3);
    v4f* o0 = reinterpret_cast<v4f*>(out + s0);
    v4f* o1 = reinterpret_cast<v4f*>(out + s1);
    v4f* o2 = reinterpret_cast<v4f*>(out + s2);
    v4f* o3 = reinterpret_cast<v4f*>(out + s3);

    for (int i = tid; i < THW4; i += 256) {
        const v4f v0 = x0[i], v1 = x1[i], v2 = x2[i], v3 = x3[i];
        v4f r0 = {}, r1 = {}, r2 = {}, r3 = {};
        r0 = f4fma(k1, v0, r0); r0 = f4fma(k2, v1, r0);
        r1 = f4fma(k0, v0, r1); r1 = f4fma(k1, v1, r1); r1 = f4fma(k2, v2, r1);
        r2 = f4fma(k0, v1, r2); r2 = f4fma(k1, v2, r2); r2 = f4fma(k2, v3, r2);
        r3 = f4fma(k0, v2, r3); r3 = f4fma(k1, v3, r3);
        __builtin_nontemporal_store(r0, &o0[i]);
        __builtin_nontemporal_store(r1, &o1[i]);
        __builtin_nontemporal_store(r2, &o2[i]);
        __builtin_nontemporal_store(r3, &o3[i]);
    }
}

// ---------------------------------------------------------------------------
extern "C" void kernel_launch(void* const* d_in, const int* in_sizes, int n_in,
                              void* d_out, int out_size, void* d_ws, size_t ws_size,
                              hipStream_t stream) {
    const float* x     = (const float*)d_in[0];
    const float* W1    = (const float*)d_in[1];
    const float* gamma = (const float*)d_in[2];
    const float* beta  = (const float*)d_in[3];
    const float* mean  = (const float*)d_in[4];
    const float* var   = (const float*)d_in[5];
    const float* W2    = (const float*)d_in[6];
    float* out = (float*)d_out;

    float* pooled = (float*)d_ws;        // R*U floats (32 KB)
    float* kern   = pooled + R * U;      // R*KK floats (24 KB)

    dsa_pool_kernel<<<NU * C, 256, 0, stream>>>(x, pooled);
    dsa_mlp_wmma_kernel<<<R / 16, 32, 0, stream>>>(pooled, W1, gamma, beta,
                                                   mean, var, W2, kern);
    dsa_conv_kernel<<<R, 256, 0, stream>>>(x, kern, out);
}